// PillarFeatureNet_24386824307260
// MI455X (gfx1250) — compile-verified
//
#include <hip/hip_runtime.h>

typedef __attribute__((ext_vector_type(2))) float v2f;
typedef __attribute__((ext_vector_type(8))) float v8f;

#define C_IN   9
#define C_OUT  64
#define NPTS   32
#define XLIM   432
#define YLIM   496
#define PLANE  (XLIM * YLIM)
#define BN_EPS 1e-3f

// ---------------------------------------------------------------------------
// Tiny per-channel BN constant precompute: inv = gamma*rsqrt(var+eps),
// bias = beta - mean*inv.  128 floats into workspace.
// ---------------------------------------------------------------------------
__global__ void bn_precompute_kernel(const float* __restrict__ gamma,
                                     const float* __restrict__ beta,
                                     const float* __restrict__ mean,
                                     const float* __restrict__ var,
                                     float* __restrict__ bn) {
    int c = threadIdx.x;
    if (c < C_OUT) {
        float inv = gamma[c] * rsqrtf(var[c] + BN_EPS);
        bn[c]        = inv;
        bn[C_OUT + c] = beta[c] - mean[c] * inv;
    }
}

// ---------------------------------------------------------------------------
// Fill cell->pillar index map with -1 (vectorized int4 stores).
// ---------------------------------------------------------------------------
__global__ void idx_init_kernel(int4* __restrict__ idxmap4, int n4) {
    int i = blockIdx.x * blockDim.x + threadIdx.x;
    if (i < n4) idxmap4[i] = make_int4(-1, -1, -1, -1);
}

// ---------------------------------------------------------------------------
// Scatter pillar index into the (b, y, x) map (inverts the reference scatter).
// ---------------------------------------------------------------------------
__global__ void idx_scatter_kernel(const int* __restrict__ coors,
                                   int* __restrict__ idxmap, int P) {
    int p = blockIdx.x * blockDim.x + threadIdx.x;
    if (p < P) {
        int b = coors[3 * p + 0];
        int x = coors[3 * p + 1];
        int y = coors[3 * p + 2];
        idxmap[(b * YLIM + y) * XLIM + x] = p;
    }
}

// ---------------------------------------------------------------------------
// One wave per pillar (grid-stride).  D = A x B with
//   A = features tile  (M = 16 points, K = 4 of C_IN)   -> 2 M-tiles
//   B = conv_w tile    (K = 4 of C_IN, N = 16 channels) -> 4 N-tiles
// K = 9 padded to 12 -> 3 WMMA k-steps; 2*4*3 = 24 v_wmma per pillar.
// D layout: lane holds channel N = (lane&15), 8 regs = 8 points -> the
// 32-point max/min reduction is in-lane v_max/v_min plus one shfl_xor(16).
// BN (exact for any sign of inv) + ReLU applied to the reduced pair.
// ---------------------------------------------------------------------------
__global__ void __launch_bounds__(256) pillar_wmma_kernel(
    const float* __restrict__ feat, const float* __restrict__ w,
    const float* __restrict__ bn, float* __restrict__ pooled, int P) {

    const int lane = threadIdx.x & 31;
    const int lh   = lane & 15;      // N (channel within tile) / M (point within tile)
    const int hi   = lane >> 4;      // upper half-wave -> K offset +2, M offset +8
    const int wave = (blockIdx.x * blockDim.x + threadIdx.x) >> 5;
    const int nwav = (gridDim.x * blockDim.x) >> 5;

    // Weight B-tiles: lane lh = channel within tile, VGPR pair = K {k0, k0+1},
    // hi half supplies K+2.  Loaded once per wave, reused over all pillars.
    v2f Bw[4][3];
#pragma unroll
    for (int nt = 0; nt < 4; ++nt) {
        const int ch = nt * 16 + lh;
#pragma unroll
        for (int ks = 0; ks < 3; ++ks) {
            const int k0 = ks * 4 + hi * 2;
            const int k1 = k0 + 1;
            float b0 = w[ch * C_IN + (k0 < C_IN ? k0 : C_IN - 1)];
            float b1 = w[ch * C_IN + (k1 < C_IN ? k1 : C_IN - 1)];
            v2f t;
            t.x = (k0 < C_IN) ? b0 : 0.0f;
            t.y = (k1 < C_IN) ? b1 : 0.0f;
            Bw[nt][ks] = t;
        }
    }

    // Per-lane BN constants: channel = nt*16 + lh (same for all 8 D regs).
    float bnS[4], bnB[4];
#pragma unroll
    for (int nt = 0; nt < 4; ++nt) {
        bnS[nt] = bn[nt * 16 + lh];
        bnB[nt] = bn[C_OUT + nt * 16 + lh];
    }

    for (int p = wave; p < P; p += nwav) {
        const float* f = feat + (size_t)p * (C_IN * NPTS);

        // Feature A-tiles: lane lh = point within tile, VGPR pair = K.
        v2f Af[2][3];
#pragma unroll
        for (int mt = 0; mt < 2; ++mt) {
            const int pt = mt * 16 + lh;
#pragma unroll
            for (int ks = 0; ks < 3; ++ks) {
                const int k0 = ks * 4 + hi * 2;
                const int k1 = k0 + 1;
                float a0 = f[(k0 < C_IN ? k0 : C_IN - 1) * NPTS + pt];
                float a1 = f[(k1 < C_IN ? k1 : C_IN - 1) * NPTS + pt];
                v2f t;
                t.x = (k0 < C_IN) ? a0 : 0.0f;
                t.y = (k1 < C_IN) ? a1 : 0.0f;
                Af[mt][ks] = t;
            }
        }

#pragma unroll
        for (int nt = 0; nt < 4; ++nt) {
            v8f acc0 = {0.f, 0.f, 0.f, 0.f, 0.f, 0.f, 0.f, 0.f};  // points 0-15
            v8f acc1 = {0.f, 0.f, 0.f, 0.f, 0.f, 0.f, 0.f, 0.f};  // points 16-31
#pragma unroll
            for (int ks = 0; ks < 3; ++ks) {
                acc0 = __builtin_amdgcn_wmma_f32_16x16x4_f32(
                    false, Af[0][ks], false, Bw[nt][ks], (short)0, acc0, false, false);
                acc1 = __builtin_amdgcn_wmma_f32_16x16x4_f32(
                    false, Af[1][ks], false, Bw[nt][ks], (short)0, acc1, false, false);
            }
            // in-lane reduce over 16 points, then fold the other half-wave
            float mx = acc0[0], mn = acc0[0];
#pragma unroll
            for (int r = 1; r < 8; ++r) { mx = fmaxf(mx, acc0[r]); mn = fminf(mn, acc0[r]); }
#pragma unroll
            for (int r = 0; r < 8; ++r) { mx = fmaxf(mx, acc1[r]); mn = fminf(mn, acc1[r]); }
            mx = fmaxf(mx, __shfl_xor(mx, 16, 32));
            mn = fminf(mn, __shfl_xor(mn, 16, 32));

            // max_n relu(s*x+b) == max(relu(s*mx+b), relu(s*mn+b)) for any sign of s
            const float s = bnS[nt], o = bnB[nt];
            float res = fmaxf(fmaxf(s * mx + o, 0.0f), fmaxf(s * mn + o, 0.0f));

            if (hi == 0)
                pooled[(size_t)p * C_OUT + nt * 16 + lh] = res;  // 16-lane coalesced
        }
    }
}

// ---------------------------------------------------------------------------
// Gather pass: every output element written once, fully coalesced (float4).
// idxmap (6.9MB) and pooled (33MB) are L2-resident on MI455X (192MB L2).
// out[b][c][y][x] = idx>=0 ? pooled[idx][c] : 0
// ---------------------------------------------------------------------------
__global__ void gather_kernel(const int4* __restrict__ idxmap4,
                              const float* __restrict__ pooled,
                              float4* __restrict__ out4, int total4) {
    int i = blockIdx.x * blockDim.x + threadIdx.x;
    if (i >= total4) return;
    int oi = i * 4;                 // flat = (b*64+c)*PLANE + y*432 + x
    int sp = oi % PLANE;            // y*432 + x   (x % 4 == 0)
    int bc = oi / PLANE;
    int b = bc >> 6;
    int c = bc & 63;
    int4 idx = idxmap4[(b * PLANE + sp) >> 2];
    float4 v;
    v.x = (idx.x >= 0) ? pooled[(size_t)idx.x * C_OUT + c] : 0.0f;
    v.y = (idx.y >= 0) ? pooled[(size_t)idx.y * C_OUT + c] : 0.0f;
    v.z = (idx.z >= 0) ? pooled[(size_t)idx.z * C_OUT + c] : 0.0f;
    v.w = (idx.w >= 0) ? pooled[(size_t)idx.w * C_OUT + c] : 0.0f;
    out4[i] = v;
}

// ---------------------------------------------------------------------------
extern "C" void kernel_launch(void* const* d_in, const int* in_sizes, int n_in,
                              void* d_out, int out_size, void* d_ws, size_t ws_size,
                              hipStream_t stream) {
    const float* feat  = (const float*)d_in[0];
    const float* w     = (const float*)d_in[1];
    const float* gamma = (const float*)d_in[2];
    const float* beta  = (const float*)d_in[3];
    const float* mean  = (const float*)d_in[4];
    const float* var   = (const float*)d_in[5];
    const int*   coors = (const int*)d_in[6];
    float* out = (float*)d_out;

    const int P  = in_sizes[0] / (C_IN * NPTS);          // 128000
    const int nb = out_size / (C_OUT * PLANE);           // 8 batches
    const int nIdx  = nb * PLANE;                        // map cells
    const int nIdx4 = nIdx / 4;
    const int total4 = out_size / 4;

    // workspace layout: pooled | idxmap | bn
    float* pooled = (float*)d_ws;
    int*   idxmap = (int*)((char*)d_ws + (size_t)P * C_OUT * sizeof(float));
    float* bnbuf  = (float*)((char*)idxmap + (size_t)nIdx * sizeof(int));

    bn_precompute_kernel<<<1, 64, 0, stream>>>(gamma, beta, mean, var, bnbuf);
    idx_init_kernel<<<(nIdx4 + 255) / 256, 256, 0, stream>>>((int4*)idxmap, nIdx4);
    idx_scatter_kernel<<<(P + 255) / 256, 256, 0, stream>>>(coors, idxmap, P);
    pillar_wmma_kernel<<<1024, 256, 0, stream>>>(feat, w, bnbuf, pooled, P);
    gather_kernel<<<(total4 + 255) / 256, 256, 0, stream>>>(
        (const int4*)idxmap, pooled, (float4*)out, total4);
}